// LinearLayer_53369263620497
// MI455X (gfx1250) — compile-verified
//
#include <hip/hip_runtime.h>

// ---------------------------------------------------------------------------
// Hypernetwork: out[b,o] = sum_i W[b,o,i] x[b,i] + bias[b,o]
// where W[b] = sum_k h[b,k]*G_k + G_bias  (G_k = gen_W[:,k] reshaped OUTxIN)
// Factorized as one GEMM: A[b,(k,i)] = h'[b,k]*x[b,i]   (h'[b,8] = 1)
//                          Wbig[(k,i),o] = Wt[k][i][o]
// out = A @ Wbig + hbias,  M=4096 N=128 K=1152, fp32 via V_WMMA_F32_16X16X4_F32
// ---------------------------------------------------------------------------

#define B_ROWS 4096
#define IN_F   128
#define OUT_F  128
#define NW     (IN_F * OUT_F)
#define E0     10
#define E1     16
#define E2     8
#define KMATS  9          // 8 h-dims + 1 constant slot for gen_b weights
#define TB     256
#define PAD    4
#define BS_I   64         // i-rows of one B slab staged in LDS at a time

typedef __attribute__((ext_vector_type(2))) float v2f;
typedef __attribute__((ext_vector_type(8))) float v8f;

// ---- pre-kernel: Wt[k][i][o] = (k<8) ? gen_W[(o*IN+i)*8 + k] : gen_b[o*IN+i]
__global__ __launch_bounds__(TB) void build_wt(const float* __restrict__ gen_W,
                                               const float* __restrict__ gen_b,
                                               float* __restrict__ Wt) {
  int idx = blockIdx.x * TB + threadIdx.x;          // [0, 9*128*128)
  if (idx >= KMATS * IN_F * OUT_F) return;
  int o = idx & (OUT_F - 1);
  int i = (idx >> 7) & (IN_F - 1);
  int k = idx >> 14;
  int row = o * IN_F + i;
  Wt[idx] = (k < E2) ? gen_W[row * E2 + k] : gen_b[row];   // write coalesced
}

// ---- fused MLP + WMMA GEMM + bias epilogue ----
__global__ __launch_bounds__(TB) void hyper_gemm(
    const float* __restrict__ x,        const float* __restrict__ features,
    const float* __restrict__ fc0_W,    const float* __restrict__ fc0_b,
    const float* __restrict__ prelu0_a, const float* __restrict__ fc1_W,
    const float* __restrict__ fc1_b,    const float* __restrict__ prelu1_a,
    const float* __restrict__ gen_W,    const float* __restrict__ gen_b,
    const float* __restrict__ Wt,       int use_wt,
    float* __restrict__ out) {
  __shared__ float Xs[16][IN_F + PAD];                       // 8.4 KB
  __shared__ float Hs[16][KMATS];                            // tiny
  __shared__ __align__(16) float Bs[BS_I][OUT_F + PAD];      // 33.8 KB

  const int tid     = threadIdx.x;
  const int wave    = tid >> 5;
  const int lane    = tid & 31;
  const int rowBase = blockIdx.x * 16;

  // stage x tile (16 rows x 128)
  for (int e = tid; e < 16 * IN_F; e += TB) {
    int m = e >> 7, i = e & 127;
    Xs[m][i] = x[(size_t)(rowBase + m) * IN_F + i];
  }

  // tiny per-row MLP: threads 0..15 each own one batch row
  if (tid < 16) {
    int b = rowBase + tid;
    float f[E0];
#pragma unroll
    for (int j = 0; j < E0; ++j) f[j] = features[b * E0 + j];
    float a0 = prelu0_a[0], a1 = prelu1_a[0];
    float h1[E1];
#pragma unroll
    for (int r = 0; r < E1; ++r) {
      float acc = fc0_b[r];
#pragma unroll
      for (int j = 0; j < E0; ++j) acc = __builtin_fmaf(fc0_W[r * E0 + j], f[j], acc);
      h1[r] = acc >= 0.f ? acc : a0 * acc;
    }
#pragma unroll
    for (int r = 0; r < E2; ++r) {
      float acc = fc1_b[r];
#pragma unroll
      for (int j = 0; j < E1; ++j) acc = __builtin_fmaf(fc1_W[r * E1 + j], h1[j], acc);
      Hs[tid][r] = acc >= 0.f ? acc : a1 * acc;
    }
    Hs[tid][E2] = 1.0f;                       // constant slot -> gen_b weights
  }
  __syncthreads();

  // WMMA fragment index helpers (ISA 7.12.2 layouts, wave32)
  const int mrow  = lane & 15;                // A rows / Hs row
  const int hi    = lane >> 4;                // lane half: K offset +2 (A/B)
  const int nb    = wave * 16;                // this wave's N tile
  const int nlane = nb + (lane & 15);         // C/B column this lane owns

  v8f acc = (v8f){0.f, 0.f, 0.f, 0.f, 0.f, 0.f, 0.f, 0.f};

  for (int k = 0; k < KMATS; ++k) {
    float hk = Hs[mrow][k];
    for (int half = 0; half < 2; ++half) {
      const int ibase = half * BS_I;
      __syncthreads();                        // waves done reading previous Bs
      if (use_wt) {                           // coalesced float4 stage from Wt
        const float4* src =
            (const float4*)(Wt + ((size_t)k * IN_F + ibase) * OUT_F);
        for (int e = tid; e < BS_I * (OUT_F / 4); e += TB) {
          int ii = e >> 5, c4 = (e & 31) << 2;
          *(float4*)&Bs[ii][c4] = src[e];
        }
      } else {                                // direct scattered (L2-resident)
        for (int e = tid; e < BS_I * OUT_F; e += TB) {
          int ii = e >> 7, o = e & 127;
          int row = o * IN_F + (ibase + ii);
          Bs[ii][o] = (k < E2) ? gen_W[row * E2 + k] : gen_b[row];
        }
      }
      __syncthreads();

#pragma unroll
      for (int kk = 0; kk < BS_I; kk += 4) {
        const int ka = kk + (hi << 1);        // this lane's K pair base
        v2f a, bfrag;
        a.x = hk * Xs[mrow][ibase + ka];      // A[b,(k,i)] = h[b,k]*x[b,i]
        a.y = hk * Xs[mrow][ibase + ka + 1];
        bfrag.x = Bs[ka][nlane];
        bfrag.y = Bs[ka + 1][nlane];
        acc = __builtin_amdgcn_wmma_f32_16x16x4_f32(
            /*neg_a=*/false, a, /*neg_b=*/false, bfrag,
            /*c_mod=*/(short)0, acc, /*reuse_a=*/false, /*reuse_b=*/false);
      }
    }
  }

  // epilogue: generated bias  hbias[b,o] = gen_b[NW+o] + sum_k h[b,k]*gen_W[NW+o,k]
  float gw[E2];
#pragma unroll
  for (int k2 = 0; k2 < E2; ++k2) gw[k2] = gen_W[(NW + nlane) * E2 + k2];
  const float gb = gen_b[NW + nlane];

#pragma unroll
  for (int r = 0; r < 8; ++r) {               // C layout: vgpr r -> M = r + hi*8
    const int m = r + (hi << 3);
    float hb = gb;
#pragma unroll
    for (int k2 = 0; k2 < E2; ++k2) hb = __builtin_fmaf(Hs[m][k2], gw[k2], hb);
    out[(size_t)(rowBase + m) * OUT_F + nlane] = acc[r] + hb;
  }
}

extern "C" void kernel_launch(void* const* d_in, const int* in_sizes, int n_in,
                              void* d_out, int out_size, void* d_ws, size_t ws_size,
                              hipStream_t stream) {
  const float* x        = (const float*)d_in[0];
  const float* features = (const float*)d_in[1];
  const float* fc0_W    = (const float*)d_in[2];
  const float* fc0_b    = (const float*)d_in[3];
  const float* prelu0_a = (const float*)d_in[4];
  const float* fc1_W    = (const float*)d_in[5];
  const float* fc1_b    = (const float*)d_in[6];
  const float* prelu1_a = (const float*)d_in[7];
  const float* gen_W    = (const float*)d_in[8];
  const float* gen_b    = (const float*)d_in[9];
  float* out = (float*)d_out;

  const size_t wt_bytes = (size_t)KMATS * IN_F * OUT_F * sizeof(float); // 576 KB
  float* Wt = (float*)d_ws;
  const int use_wt = (d_ws != nullptr && ws_size >= wt_bytes) ? 1 : 0;

  if (use_wt) {
    const int n = KMATS * IN_F * OUT_F;
    build_wt<<<(n + TB - 1) / TB, TB, 0, stream>>>(gen_W, gen_b, Wt);
  }
  hyper_gemm<<<B_ROWS / 16, TB, 0, stream>>>(
      x, features, fc0_W, fc0_b, prelu0_a, fc1_W, fc1_b, prelu1_a,
      gen_W, gen_b, Wt, use_wt, out);
}